// SimpleCompressedAttention_57019985822209
// MI455X (gfx1250) — compile-verified
//
#include <hip/hip_runtime.h>
#include <hip/hip_bf16.h>
#include <cmath>

// ---------------- problem constants ----------------
constexpr int BSZ = 16;
constexpr int NH  = 32;      // heads
constexpr int HD  = 128;     // head dim
constexpr int LR  = 512;     // lora rank (latent dim)
constexpr int QLR = 1536;    // q lora rank
constexpr int SL  = 4096;    // seq len
constexpr int HDTOT = NH * HD;      // 4096

// flash decode tiling
constexpr int CHUNKS = 16;
constexpr int SC = SL / CHUNKS;     // 256 positions per chunk
constexpr int ST = 16;              // s-tile per iteration
constexpr int NIT = SC / ST;        // 16 iterations

// padded LDS strides (floats) to avoid bank conflicts on strided fragment reads
constexpr int QSP = LR + 4;   // 516 -> 2064B rows, 16B aligned, banks spread by 4
constexpr int CSP = LR + 4;   // 516
constexpr int PSP = 20;       // P matrix row stride
constexpr int CSTILE = ST * CSP;

typedef float v2f __attribute__((ext_vector_type(2)));
typedef float v8f __attribute__((ext_vector_type(8)));
typedef int   v4i __attribute__((ext_vector_type(4)));

static __device__ __forceinline__ v8f wmma_f32x4(v2f a, v2f b, v8f c) {
  // D = A(16x4 f32) * B(4x16 f32) + C(16x16 f32)
  return __builtin_amdgcn_wmma_f32_16x16x4_f32(
      /*neg_a=*/false, a, /*neg_b=*/false, b,
      /*c_mod=*/(short)0, c, /*reuse_a=*/false, /*reuse_b=*/false);
}

// ---------------- CDNA5 async global->LDS copy (ASYNCcnt tracked) ----------------
#define USE_ASYNC 0
#if defined(__has_builtin)
#if __has_builtin(__builtin_amdgcn_global_load_async_to_lds_b128) && \
    __has_builtin(__builtin_amdgcn_s_wait_asynccnt)
#undef USE_ASYNC
#define USE_ASYNC 1
#endif
#endif

typedef __attribute__((address_space(1))) v4i gvec4;   // global int4
typedef __attribute__((address_space(3))) v4i lvec4;   // LDS int4

static __device__ __forceinline__ void copy16_to_lds(const float* g, float* l) {
#if USE_ASYNC
  __builtin_amdgcn_global_load_async_to_lds_b128(
      (gvec4*)(uintptr_t)g, (lvec4*)(unsigned)(uintptr_t)l, 0, 0);
#else
  *(float4*)l = *(const float4*)g;
#endif
}

static __device__ __forceinline__ void wait_async0() {
#if USE_ASYNC
  __builtin_amdgcn_s_wait_asynccnt(0);
#endif
}

// ---------------- kernel 1: q = hidden @ Wq^T + bq ----------------
__global__ __launch_bounds__(256) void qproj_kernel(
    const float* __restrict__ hs, const float* __restrict__ Wq,
    const float* __restrict__ bq, float* __restrict__ qout) {
  const int b = blockIdx.y;
  const int n = blockIdx.x * 256 + threadIdx.x;
  __shared__ __align__(16) float h_s[QLR];
  for (int i = threadIdx.x; i < QLR; i += 256) h_s[i] = hs[b * QLR + i];
  __syncthreads();
  const float4* w = (const float4*)(Wq + (size_t)n * QLR);
  float acc = bq[n];
#pragma unroll 4
  for (int k = 0; k < QLR / 4; ++k) {
    float4 wv = w[k];
    float4 hv = *(const float4*)&h_s[4 * k];
    acc += wv.x * hv.x + wv.y * hv.y + wv.z * hv.z + wv.w * hv.w;
  }
  qout[(size_t)b * HDTOT + n] = acc;
}

// ---------------- kernel 2: absorb q into latent space ----------------
__global__ __launch_bounds__(256) void qabsorb_kernel(
    const float* __restrict__ q, const float* __restrict__ Wkv,
    const float* __restrict__ bkv, float* __restrict__ qabs,
    float* __restrict__ qb) {
  const int b = blockIdx.y, h = blockIdx.x;
  __shared__ float qrow[HD];
  if (threadIdx.x < HD) qrow[threadIdx.x] = q[(size_t)b * HDTOT + h * HD + threadIdx.x];
  __syncthreads();
#pragma unroll
  for (int rr = 0; rr < 2; ++rr) {
    const int r = threadIdx.x + rr * 256;
    const float* wcol = Wkv + (size_t)h * HD * LR + r;
    float acc = 0.f;
#pragma unroll 8
    for (int d = 0; d < HD; ++d) acc += qrow[d] * wcol[(size_t)d * LR];
    qabs[((size_t)b * NH + h) * LR + r] = acc;
  }
  if (threadIdx.x == 0) {
    float acc = 0.f;
    for (int d = 0; d < HD; ++d) acc += qrow[d] * bkv[h * HD + d];
    qb[b * NH + h] = acc;
  }
}

// ---------------- kernel 3: flash-decode over compressed KV ----------------
// grid (16 chunks, 16 b), block 256 (8 waves).
// Balanced: every wave = (h-tile, K-quarter) for scores (32 wmma each),
// partials reduced in LDS by waves 0/1 (online softmax), then all waves
// accumulate ctx[32x512] (32 wmma each). c-tiles double-buffered with
// async global->LDS prefetch overlapping both WMMA phases.
__global__ __launch_bounds__(256) void flash_kernel(
    const float* __restrict__ ckv, const float* __restrict__ qabs,
    const float* __restrict__ qb, float* __restrict__ ctx_part,
    float* __restrict__ ml_part) {
  const int b = blockIdx.y, chunk = blockIdx.x;
  const int tid = threadIdx.x;
  const int wave = tid >> 5, lane = tid & 31;
  const int m16 = lane & 15, hi = lane >> 4, kup = hi * 2;
  const int ht = wave & 1, h0 = ht * 16;   // score h-tile for this wave
  const int kq = wave >> 1;                // score K-quarter (128 wide)
  const int rb = wave >> 1;                // ctx r-block (128 wide)

  __shared__ __align__(16) float qs[NH * QSP];       // 66 KB
  __shared__ __align__(16) float cs[2 * CSTILE];     // 66 KB (double buffered)
  __shared__ float sred[8][16][17];                  // score partials, 8.5 KB
  __shared__ float ps[NH * PSP];                     // P (post-softmax)
  __shared__ float mrun[NH], lrun[NH], alphas[NH], qbs[NH];

  // stage q_abs for this batch
  const float* qa = qabs + (size_t)b * NH * LR;
  for (int i = tid; i < NH * LR; i += 256) {
    const int h = i >> 9, r = i & (LR - 1);
    qs[h * QSP + r] = qa[i];
  }
  if (tid < NH) {
    qbs[tid] = qb[b * NH + tid];
    mrun[tid] = -INFINITY;
    lrun[tid] = 0.f;
  }

  v8f acc[8];
  const v8f vzero = {0.f, 0.f, 0.f, 0.f, 0.f, 0.f, 0.f, 0.f};
#pragma unroll
  for (int rt = 0; rt < 8; ++rt) acc[rt] = vzero;

  const float* cbase = ckv + ((size_t)b * SL + (size_t)chunk * SC) * LR;

  // prefetch tile 0 into buffer 0
#pragma unroll
  for (int i = 0; i < 8; ++i) {
    const int idx = tid + i * 256;
    const int row = idx >> 7, c4 = idx & 127;
    copy16_to_lds(cbase + row * LR + c4 * 4, &cs[row * CSP + c4 * 4]);
  }

  for (int it = 0; it < NIT; ++it) {
    const float* cs_cur = &cs[(it & 1) * CSTILE];

    wait_async0();        // current buffer fully landed in LDS
    __syncthreads();      // B0: also fences qs/init on it==0, sred/ps reuse

    // issue async prefetch of next tile into the other buffer
    if (it + 1 < NIT) {
      const float* csrc = cbase + (size_t)(it + 1) * ST * LR;
      float* dst = &cs[((it + 1) & 1) * CSTILE];
#pragma unroll
      for (int i = 0; i < 8; ++i) {
        const int idx = tid + i * 256;
        const int row = idx >> 7, c4 = idx & 127;
        copy16_to_lds(csrc + row * LR + c4 * 4, &dst[row * CSP + c4 * 4]);
      }
    }

    // ---- score partials: S_part[16h x 16s] over K in [kq*128, kq*128+128) ----
    {
      v8f sc = vzero;
      const float* qp = &qs[(h0 + m16) * QSP + kq * 128 + kup];  // A: q_abs[h, k..]
      const float* cp = &cs_cur[m16 * CSP + kq * 128 + kup];     // B: c[s, k..]
#pragma unroll 8
      for (int k = 0; k < 128; k += 4) {
        v2f A, B;
        A[0] = qp[k];  A[1] = qp[k + 1];
        B[0] = cp[k];  B[1] = cp[k + 1];
        sc = wmma_f32x4(A, B, sc);
      }
#pragma unroll
      for (int v = 0; v < 8; ++v) sred[wave][v + 8 * hi][m16] = sc[v];
    }
    __syncthreads();      // B1: partials visible

    // ---- reduce partials + online softmax (waves 0 and 1) ----
    if (wave < 2) {
#pragma unroll
      for (int v = 0; v < 8; ++v) {
        const int h = h0 + v + 8 * hi;
        const int M = v + 8 * hi;
        const float s = qbs[h] + sred[ht][M][m16] + sred[2 + ht][M][m16] +
                        sred[4 + ht][M][m16] + sred[6 + ht][M][m16];
        float r = s;
        r = fmaxf(r, __shfl_xor(r, 1));
        r = fmaxf(r, __shfl_xor(r, 2));
        r = fmaxf(r, __shfl_xor(r, 4));
        r = fmaxf(r, __shfl_xor(r, 8));          // row max over 16 s
        const float mo = mrun[h];
        const float mn = fmaxf(mo, r);
        const float al = __expf(mo - mn);        // exp(-inf)=0 on first tile
        const float p  = __expf(s - mn);
        float ls = p;
        ls += __shfl_xor(ls, 1);
        ls += __shfl_xor(ls, 2);
        ls += __shfl_xor(ls, 4);
        ls += __shfl_xor(ls, 8);
        if (m16 == 0) {
          mrun[h] = mn;
          lrun[h] = lrun[h] * al + ls;
          alphas[h] = al;
        }
        ps[h * PSP + m16] = p;
      }
    }
    __syncthreads();      // B2: ps/alphas ready

    // ---- ctx accumulation (all 8 waves): rescale then ctx += P * c ----
#pragma unroll
    for (int v = 0; v < 8; ++v) {
      const float al = alphas[h0 + v + 8 * hi];
#pragma unroll
      for (int rt = 0; rt < 8; ++rt) acc[rt][v] *= al;
    }
    const float* pp = &ps[(h0 + m16) * PSP + kup];     // A: P[h, s]
#pragma unroll
    for (int rt = 0; rt < 8; ++rt) {
      const int n0c = rb * 128 + rt * 16;
      const float* cbp = &cs_cur[kup * CSP + n0c + m16];  // B: c[s, r]
      v8f a = acc[rt];
#pragma unroll
      for (int k = 0; k < ST; k += 4) {
        v2f A, B;
        A[0] = pp[k];            A[1] = pp[k + 1];
        B[0] = cbp[k * CSP];     B[1] = cbp[(k + 1) * CSP];
        a = wmma_f32x4(A, B, a);
      }
      acc[rt] = a;
    }
    // next iteration's B0 barrier separates these reads from buffer reuse
  }

  // ---- write per-chunk partials ----
  float* cp_out = ctx_part + ((size_t)b * CHUNKS + chunk) * NH * LR;
#pragma unroll
  for (int rt = 0; rt < 8; ++rt) {
    const int rcol = rb * 128 + rt * 16 + m16;
#pragma unroll
    for (int v = 0; v < 8; ++v) {
      const int h = h0 + v + 8 * hi;
      cp_out[(size_t)h * LR + rcol] = acc[rt][v];
    }
  }
  __syncthreads();   // mrun/lrun final (written before B2 of last iter)
  if (tid < NH) {
    float2 ml = {mrun[tid], lrun[tid]};
    *(float2*)&ml_part[(((size_t)b * CHUNKS + chunk) * NH + tid) * 2] = ml;
  }
}

// ---------------- kernel 4: combine chunks + output projection ----------------
__global__ __launch_bounds__(128) void combine_kernel(
    const float* __restrict__ ctx_part, const float* __restrict__ ml_part,
    const float* __restrict__ Wkv, const float* __restrict__ bkv,
    float* __restrict__ out) {
  const int b = blockIdx.y, h = blockIdx.x, t = threadIdx.x;
  __shared__ __align__(16) float ctxs[LR];

  const float* mlb = ml_part + ((size_t)b * CHUNKS * NH + h) * 2;
  float wcs[CHUNKS];
  float M = -INFINITY;
#pragma unroll
  for (int c = 0; c < CHUNKS; ++c) M = fmaxf(M, mlb[(size_t)c * NH * 2]);
  float L = 0.f;
#pragma unroll
  for (int c = 0; c < CHUNKS; ++c) {
    wcs[c] = __expf(mlb[(size_t)c * NH * 2] - M);
    L += wcs[c] * mlb[(size_t)c * NH * 2 + 1];
  }
  const float invL = 1.f / L;

  const float* cpb = ctx_part + ((size_t)b * CHUNKS * NH + h) * LR;
  for (int r = t; r < LR; r += 128) {
    float s = 0.f;
#pragma unroll
    for (int c = 0; c < CHUNKS; ++c) s += wcs[c] * cpb[(size_t)c * NH * LR + r];
    ctxs[r] = s * invL;
  }
  __syncthreads();

  const int row = h * HD + t;
  const float4* w = (const float4*)(Wkv + (size_t)row * LR);
  float o = bkv[row];
#pragma unroll 4
  for (int k = 0; k < LR / 4; ++k) {
    float4 wv = w[k];
    float4 cv = *(const float4*)&ctxs[4 * k];
    o += wv.x * cv.x + wv.y * cv.y + wv.z * cv.z + wv.w * cv.w;
  }
  out[(size_t)b * HDTOT + row] = o;
}

// ---------------- launcher ----------------
extern "C" void kernel_launch(void* const* d_in, const int* in_sizes, int n_in,
                              void* d_out, int out_size, void* d_ws, size_t ws_size,
                              hipStream_t stream) {
  const float* hs  = (const float*)d_in[0];   // [16,1536]
  const float* ckv = (const float*)d_in[1];   // [16,4096,512]
  const float* Wq  = (const float*)d_in[2];   // [4096,1536]
  const float* bq  = (const float*)d_in[3];   // [4096]
  const float* Wkv = (const float*)d_in[4];   // [4096,512]
  const float* bkv = (const float*)d_in[5];   // [4096]
  float* out = (float*)d_out;                 // [16,4096]

  float* ws = (float*)d_ws;
  float* q        = ws;                                      // 65536
  float* qabs     = q + (size_t)BSZ * HDTOT;                 // 262144
  float* qb       = qabs + (size_t)BSZ * NH * LR;            // 512
  float* ml_part  = qb + (size_t)BSZ * NH;                   // 16384
  float* ctx_part = ml_part + (size_t)BSZ * CHUNKS * NH * 2; // 4194304

  qproj_kernel<<<dim3(HDTOT / 256, BSZ), 256, 0, stream>>>(hs, Wq, bq, q);
  qabsorb_kernel<<<dim3(NH, BSZ), 256, 0, stream>>>(q, Wkv, bkv, qabs, qb);
  flash_kernel<<<dim3(CHUNKS, BSZ), 256, 0, stream>>>(ckv, qabs, qb, ctx_part, ml_part);
  combine_kernel<<<dim3(NH, BSZ), 128, 0, stream>>>(ctx_part, ml_part, Wkv, bkv, out);
}